// SAILRNET_35673998360991
// MI455X (gfx1250) — compile-verified
//
#include <hip/hip_runtime.h>
#include <math.h>

typedef __attribute__((ext_vector_type(2))) float v2f;
typedef __attribute__((ext_vector_type(8))) float v8f;

__device__ __forceinline__ v8f wmma4(v2f a, v2f b, v8f c) {
    return __builtin_amdgcn_wmma_f32_16x16x4_f32(false, a, false, b, (short)0, c,
                                                 false, false);
}

// ---------------- wave helpers (wave32) ----------------
__device__ inline float waveRedSum(float v) {
    v += __shfl_xor(v, 16, 32);
    v += __shfl_xor(v, 8, 32);
    v += __shfl_xor(v, 4, 32);
    v += __shfl_xor(v, 2, 32);
    v += __shfl_xor(v, 1, 32);
    return v;
}
__device__ inline float waveRedMax(float v) {
    v = fmaxf(v, __shfl_xor(v, 16, 32));
    v = fmaxf(v, __shfl_xor(v, 8, 32));
    v = fmaxf(v, __shfl_xor(v, 4, 32));
    v = fmaxf(v, __shfl_xor(v, 2, 32));
    v = fmaxf(v, __shfl_xor(v, 1, 32));
    return v;
}
__device__ inline float halfRedSum(float v) {  // sum within 16-lane groups
    v += __shfl_xor(v, 8, 16);
    v += __shfl_xor(v, 4, 16);
    v += __shfl_xor(v, 2, 16);
    v += __shfl_xor(v, 1, 16);
    return v;
}

// ---------------- misc kernels ----------------
__global__ void k_zero(float* p, int n) {
    int i = blockIdx.x * blockDim.x + threadIdx.x;
    if (i < n) p[i] = 0.f;
}

// LN(pos_table[sp_x[b]]) + LN(pos_table[sp_y[b]]) -> out[b][128]; one wave per b
__global__ void k_pos_ln(const int* __restrict__ spx, const int* __restrict__ spy,
                         const float* __restrict__ table, const float* __restrict__ g,
                         const float* __restrict__ beta, float* __restrict__ out) {
    int b = blockIdx.x * (blockDim.x >> 5) + (threadIdx.x >> 5);
    if (b >= 32) return;
    int lane = threadIdx.x & 31;
    float acc[4] = {0.f, 0.f, 0.f, 0.f};
    for (int which = 0; which < 2; ++which) {
        int idx = which ? spy[b] : spx[b];
        const float* src = table + (size_t)idx * 128;
        float x[4]; float s = 0.f;
        for (int j = 0; j < 4; ++j) { x[j] = src[lane + 32 * j]; s += x[j]; }
        float m = waveRedSum(s) * (1.f / 128.f);
        float vs = 0.f;
        for (int j = 0; j < 4; ++j) { float d = x[j] - m; vs += d * d; }
        float var = waveRedSum(vs) * (1.f / 128.f);
        float inv = rsqrtf(var + 1e-5f);
        for (int j = 0; j < 4; ++j) {
            int d = lane + 32 * j;
            acc[j] += (x[j] - m) * inv * g[d] + beta[d];
        }
    }
    for (int j = 0; j < 4; ++j) out[(size_t)b * 128 + lane + 32 * j] = acc[j];
}

// emb_table gather + LayerNorm (+ optional positional add); one wave per row of 128
__global__ void k_embed_ln(const int* __restrict__ idx, const float* __restrict__ table,
                           const float* __restrict__ g, const float* __restrict__ beta,
                           const float* __restrict__ posadd, float* __restrict__ out,
                           int rows) {
    int row = blockIdx.x * (blockDim.x >> 5) + (threadIdx.x >> 5);
    if (row >= rows) return;
    int lane = threadIdx.x & 31;
    const float* src = table + (size_t)idx[row] * 128;
    float x[4]; float s = 0.f;
    for (int j = 0; j < 4; ++j) { x[j] = src[lane + 32 * j]; s += x[j]; }
    float m = waveRedSum(s) * (1.f / 128.f);
    float vs = 0.f;
    for (int j = 0; j < 4; ++j) { float d = x[j] - m; vs += d * d; }
    float var = waveRedSum(vs) * (1.f / 128.f);
    float inv = rsqrtf(var + 1e-5f);
    float* dst = out + (size_t)row * 128;
    int b = row >> 10;  // 1024 rows per batch
    for (int j = 0; j < 4; ++j) {
        int d = lane + 32 * j;
        float y = (x[j] - m) * inv * g[d] + beta[d];
        if (posadd) y += posadd[(size_t)b * 128 + d];
        dst[d] = y;
    }
}

// ---------------- WMMA fp32 GEMM: Y[M,N] = X[M,K] @ W[K,N] (+bias) -------------------
// N,K compile-time so all fragment loads use immediate offsets. One wave computes a
// 32x32 block (2x2 tiles): each A/B fragment feeds two WMMAs.
template <int N, int K>
__global__ __launch_bounds__(256) void k_gemm_wmma(
    const float* __restrict__ X, const float* __restrict__ W,
    const float* __restrict__ bias, float* __restrict__ Y, int M) {
    int wid = blockIdx.x * (blockDim.x >> 5) + (threadIdx.x >> 5);
    constexpr int NB = N >> 5;
    int nblocks = (M >> 5) * NB;
    if (wid >= nblocks) return;  // wave-uniform exit
    int m0 = (wid / NB) << 5;
    int n0 = (wid % NB) << 5;
    int lane = threadIdx.x & 31;
    int half = lane >> 4;
    int l16 = lane & 15;
    const float* a0p = X + (size_t)(m0 + l16) * K + (half << 1);
    const float* a1p = a0p + 16 * K;
    const float* b0p = W + (size_t)(half << 1) * N + n0 + l16;
    const float* b1p = b0p + 16;
    v8f c00 = {}, c01 = {}, c10 = {}, c11 = {};
    for (int k = 0; k < K; k += 4) {
        v2f a0, a1, b0, b1;
        a0.x = a0p[k];     a0.y = a0p[k + 1];
        a1.x = a1p[k];     a1.y = a1p[k + 1];
        b0.x = b0p[k * N]; b0.y = b0p[(k + 1) * N];
        b1.x = b1p[k * N]; b1.y = b1p[(k + 1) * N];
        c00 = wmma4(a0, b0, c00);
        c01 = wmma4(a0, b1, c01);
        c10 = wmma4(a1, b0, c10);
        c11 = wmma4(a1, b1, c11);
    }
    float bv0 = bias ? bias[n0 + l16] : 0.f;
    float bv1 = bias ? bias[n0 + 16 + l16] : 0.f;
    for (int r = 0; r < 8; ++r) {
        int rr = r + (half << 3);
        float* y0 = Y + (size_t)(m0 + rr) * N + n0 + l16;
        float* y1 = Y + (size_t)(m0 + 16 + rr) * N + n0 + l16;
        y0[0] = c00[r] + bv0;
        y0[16] = c01[r] + bv1;
        y1[0] = c10[r] + bv0;
        y1[16] = c11[r] + bv1;
    }
}

// ---------------- fused attention: scores -> softmax(+entropy, store attn) -> out -> pool ---
#define SROW 1032  // LDS row stride (floats); 1032 % 64 == 8 -> 2-way bank conflict on col reads
__global__ __launch_bounds__(256) void k_attention(
    const float* __restrict__ qp, const float* __restrict__ kp,
    const float* __restrict__ vp, const float* __restrict__ self_imp,
    const float* __restrict__ pool_w, float* __restrict__ attn_out,
    float* __restrict__ pool_out, float* ent_acc) {
    __shared__ float s_attn[16 * SROW];
    const int b = blockIdx.x >> 6;     // 64 row-blocks per batch
    const int rb = blockIdx.x & 63;
    const int row0 = rb << 4;
    const int w = threadIdx.x >> 5;
    const int lane = threadIdx.x & 31;
    const int half = lane >> 4, l16 = lane & 15;
    const float* Q = qp + (size_t)b * 1024 * 128;
    const float* Kp = kp + (size_t)b * 1024 * 128;
    const float* Vp = vp + (size_t)b * 1024 * 128;
    const float scale = 0.08838834764831845f;  // 1/sqrt(128)

    // Phase 1: scores rows row0..row0+15 x 1024; each wave does 4 strips of 32 cols,
    // sharing one A fragment between the two 16-col tiles of a strip.
    const float* ap = Q + (size_t)(row0 + l16) * 128 + (half << 1);
    for (int t = w; t < 32; t += 8) {
        int n0 = t << 5;
        const float* b0p = Kp + (size_t)(n0 + l16) * 128 + (half << 1);
        const float* b1p = b0p + 16 * 128;
        v8f c0 = {}, c1 = {};
        for (int k = 0; k < 128; k += 4) {
            v2f a, b0, b1;
            a.x = ap[k];   a.y = ap[k + 1];
            b0.x = b0p[k]; b0.y = b0p[k + 1];  // K^T read
            b1.x = b1p[k]; b1.y = b1p[k + 1];
            c0 = wmma4(a, b0, c0);
            c1 = wmma4(a, b1, c1);
        }
        for (int r = 0; r < 8; ++r) {
            int rl = r + (half << 3);
            int grow = row0 + rl;
            float imp = 0.f;
            int col0 = n0 + l16, col1 = n0 + 16 + l16;
            float v0 = c0[r] * scale;
            float v1 = c1[r] * scale;
            if (grow == col0 || grow == col1)
                imp = fminf(__expf(self_imp[grow]), 3.0f);
            if (grow == col0) v0 += imp;
            if (grow == col1) v1 += imp;
            s_attn[rl * SROW + col0] = v0;
            s_attn[rl * SROW + col1] = v1;
        }
    }
    __syncthreads();

    // Phase 2: softmax rows (2 per wave) + entropy + stream probs to d_out (non-temporal)
    float entp = 0.f;
    for (int rr = 0; rr < 2; ++rr) {
        int rl = (w << 1) + rr;
        float mx = -3.4e38f;
        for (int j = lane; j < 1024; j += 32) mx = fmaxf(mx, s_attn[rl * SROW + j]);
        mx = waveRedMax(mx);
        float se = 0.f;
        for (int j = lane; j < 1024; j += 32) se += __expf(s_attn[rl * SROW + j] - mx);
        se = waveRedSum(se);
        float inv = 1.f / se;
        float* grow = attn_out + ((size_t)b * 1024 + row0 + rl) * 1024;
        for (int j = lane; j < 1024; j += 32) {
            float p = __expf(s_attn[rl * SROW + j] - mx) * inv;
            s_attn[rl * SROW + j] = p;
            __builtin_nontemporal_store(p, grow + j);  // never re-read from global
            entp += p * logf(p + 1e-10f);
        }
    }
    if (ent_acc) {
        float es = waveRedSum(entp);
        if (lane == 0) atomicAdd(ent_acc, es);
    }
    __syncthreads();

    // Phase 3: out tile [16 x 16] at n0 = w*16 : attn(LDS) @ vp, then fused attn_pool
    {
        int n0 = w << 4;
        const float* bp = Vp + (size_t)(half << 1) * 128 + n0 + l16;
        v8f c = {};
        for (int k = 0; k < 1024; k += 4) {
            int kb = k + (half << 1);
            v2f a, bb;
            a.x = s_attn[l16 * SROW + kb];
            a.y = s_attn[l16 * SROW + kb + 1];
            bb.x = bp[k * 128];
            bb.y = bp[(k + 1) * 128];
            c = wmma4(a, bb, c);
        }
        // softmax(pool_w) weight for this lane's column
        float wmax = -3.4e38f;
        for (int d = 0; d < 128; ++d) wmax = fmaxf(wmax, pool_w[d]);
        float wz = 0.f;
        for (int d = 0; d < 128; ++d) wz += __expf(pool_w[d] - wmax);
        float wsm = __expf(pool_w[n0 + l16] - wmax) / wz;

        __syncthreads();  // all waves done reading s_attn; reuse LDS for pool partials
        float* part = s_attn;  // [8 waves][16 rows]
        for (int r = 0; r < 8; ++r) {
            float v = c[r] * wsm;
            v = halfRedSum(v);
            if (l16 == 0) part[w * 16 + r + (half << 3)] = v;
        }
        __syncthreads();
        if (threadIdx.x < 16) {
            float s = 0.f;
            for (int ww = 0; ww < 8; ++ww) s += part[ww * 16 + threadIdx.x];
            pool_out[(size_t)b * 1024 + row0 + threadIdx.x] = s;
        }
    }
}

// ---------------- BatchNorm (training-mode batch stats over 32 rows) + ReLU --------------
__global__ void k_bn_relu(const float* __restrict__ Y, const float* __restrict__ g,
                          const float* __restrict__ beta, float* __restrict__ H, int C) {
    int c = blockIdx.x * blockDim.x + threadIdx.x;
    if (c >= C) return;
    float s = 0.f;
    for (int r = 0; r < 32; ++r) s += Y[(size_t)r * C + c];
    float m = s * (1.f / 32.f);
    float vs = 0.f;
    for (int r = 0; r < 32; ++r) { float d = Y[(size_t)r * C + c] - m; vs += d * d; }
    float inv = rsqrtf(vs * (1.f / 32.f) + 1e-5f);
    for (int r = 0; r < 32; ++r) {
        float v = (Y[(size_t)r * C + c] - m) * inv * g[c] + beta[c];
        H[(size_t)r * C + c] = fmaxf(v, 0.f);
    }
}

// sum_b sum_d p*log(p+eps) with p = softmax(H[b,:128]) ; one wave per row
__global__ void k_row_entropy(const float* __restrict__ H, float* acc) {
    int row = blockIdx.x * (blockDim.x >> 5) + (threadIdx.x >> 5);
    if (row >= 32) return;
    int lane = threadIdx.x & 31;
    float x[4]; float mx = -3.4e38f;
    for (int j = 0; j < 4; ++j) {
        x[j] = H[(size_t)row * 128 + lane + 32 * j];
        mx = fmaxf(mx, x[j]);
    }
    mx = waveRedMax(mx);
    float se = 0.f;
    for (int j = 0; j < 4; ++j) se += __expf(x[j] - mx);
    se = waveRedSum(se);
    float ent = 0.f;
    for (int j = 0; j < 4; ++j) {
        float p = __expf(x[j] - mx) / se;
        ent += p * logf(p + 1e-10f);
    }
    ent = waveRedSum(ent);
    if (lane == 0) atomicAdd(acc, ent);
}

// acc[0]=attn plogp sum, acc[1]=h_sc plogp sum, acc[2]=h_spp plogp sum
__global__ void k_finalize(const float* __restrict__ acc, float* __restrict__ out_el) {
    float ent_hsc = -acc[1] * (1.f / 32.f);
    float ent_hspp = -acc[2] * (1.f / 32.f);
    float el_attn = -acc[0] * (1.f / 32768.f);
    *out_el = 0.1f * (ent_hsc + ent_hspp) + el_attn;
}

// ---------------- launch ----------------
extern "C" void kernel_launch(void* const* d_in, const int* in_sizes, int n_in,
                              void* d_out, int out_size, void* d_ws, size_t ws_size,
                              hipStream_t stream) {
    (void)in_sizes; (void)n_in; (void)out_size; (void)ws_size;
    const int* x_sc = (const int*)d_in[0];
    const int* x_spp = (const int*)d_in[1];
    const int* sp_x = (const int*)d_in[2];
    const int* sp_y = (const int*)d_in[3];
    const float* emb_table = (const float*)d_in[4];
    const float* emb_ln_g = (const float*)d_in[5];
    const float* emb_ln_b = (const float*)d_in[6];
    const float* pos_table = (const float*)d_in[7];
    const float* pos_ln_g = (const float*)d_in[8];
    const float* pos_ln_b = (const float*)d_in[9];
    const float* Wq = (const float*)d_in[10];
    const float* Wk = (const float*)d_in[11];
    const float* Wv = (const float*)d_in[12];
    const float* self_imp = (const float*)d_in[13];
    const float* pool_w = (const float*)d_in[14];
    const float* enc_W1 = (const float*)d_in[15];
    const float* enc_b1 = (const float*)d_in[16];
    const float* enc_g1 = (const float*)d_in[17];
    const float* enc_beta1 = (const float*)d_in[18];
    const float* enc_W2 = (const float*)d_in[19];
    const float* enc_b2 = (const float*)d_in[20];
    const float* enc_g2 = (const float*)d_in[21];
    const float* enc_beta2 = (const float*)d_in[22];
    const float* prj_W1 = (const float*)d_in[23];
    const float* prj_b1 = (const float*)d_in[24];
    const float* prj_g1 = (const float*)d_in[25];
    const float* prj_beta1 = (const float*)d_in[26];
    const float* prj_W2 = (const float*)d_in[27];
    const float* prj_b2 = (const float*)d_in[28];
    const float* prj_g2 = (const float*)d_in[29];
    const float* prj_beta2 = (const float*)d_in[30];

    float* out = (float*)d_out;
    // d_out layout (floats)
    const size_t H_SC = 0, H_SPP = 4096, Z_SC = 8192, Z_SPP = 10240;
    const size_t ATTN_SC = 12288, ATTN_SPP = ATTN_SC + 33554432ull;
    const size_t EL = ATTN_SPP + 33554432ull;

    // workspace layout (floats)
    float* ws = (float*)d_ws;
    const size_t BND = 32ull * 1024 * 128;  // 4,194,304
    float* emb_sc = ws;
    float* emb_spp = emb_sc + BND;
    float* possum = emb_spp + BND;          // [32,128]
    float* P0 = possum + 4096;              // q-proj
    float* P1 = P0 + BND;                   // k-proj
    float* P2 = P1 + BND;                   // v-proj
    float* pool_sc = P2 + BND;              // [32,1024]
    float* pool_spp = pool_sc + 32768;
    float* y1 = pool_spp + 32768;           // [32,512]
    float* h1 = y1 + 16384;
    float* y2 = h1 + 16384;                 // [32,128]
    float* y3 = y2 + 4096;                  // [32,128]
    float* h3 = y3 + 4096;
    float* y4 = h3 + 4096;                  // [32,64]
    float* acc = y4 + 2048;                 // 4 accumulator floats

    k_zero<<<1, 32, 0, stream>>>(acc, 4);

    // embeddings + layernorm
    k_pos_ln<<<4, 256, 0, stream>>>(sp_x, sp_y, pos_table, pos_ln_g, pos_ln_b, possum);
    k_embed_ln<<<4096, 256, 0, stream>>>(x_sc, emb_table, emb_ln_g, emb_ln_b,
                                         (const float*)nullptr, emb_sc, 32768);
    k_embed_ln<<<4096, 256, 0, stream>>>(x_spp, emb_table, emb_ln_g, emb_ln_b,
                                         possum, emb_spp, 32768);

    // attention 1: q=emb_sc, k=v=emb_spp  -> attn_sc, pool_sc (entropy discarded)
    k_gemm_wmma<128, 128><<<512, 256, 0, stream>>>(emb_sc, Wq, nullptr, P0, 32768);
    k_gemm_wmma<128, 128><<<512, 256, 0, stream>>>(emb_spp, Wk, nullptr, P1, 32768);
    k_gemm_wmma<128, 128><<<512, 256, 0, stream>>>(emb_spp, Wv, nullptr, P2, 32768);
    k_attention<<<2048, 256, 0, stream>>>(P0, P1, P2, self_imp, pool_w,
                                          out + ATTN_SC, pool_sc, (float*)nullptr);

    // attention 2: q=emb_spp, k=v=emb_sc -> attn_spp, pool_spp, el_attn
    k_gemm_wmma<128, 128><<<512, 256, 0, stream>>>(emb_spp, Wq, nullptr, P0, 32768);
    k_gemm_wmma<128, 128><<<512, 256, 0, stream>>>(emb_sc, Wk, nullptr, P1, 32768);
    k_gemm_wmma<128, 128><<<512, 256, 0, stream>>>(emb_sc, Wv, nullptr, P2, 32768);
    k_attention<<<2048, 256, 0, stream>>>(P0, P1, P2, self_imp, pool_w,
                                          out + ATTN_SPP, pool_spp, acc + 0);

    // encoder + projector (sc then spp, reusing temps; stream-ordered)
    const float* pools[2] = {pool_sc, pool_spp};
    const size_t hoff[2] = {H_SC, H_SPP};
    const size_t zoff[2] = {Z_SC, Z_SPP};
    for (int i = 0; i < 2; ++i) {
        // encoder: 1024 -> 512 -> 128
        k_gemm_wmma<512, 1024><<<2, 256, 0, stream>>>(pools[i], enc_W1, enc_b1, y1, 32);
        k_bn_relu<<<2, 256, 0, stream>>>(y1, enc_g1, enc_beta1, h1, 512);
        k_gemm_wmma<128, 512><<<1, 256, 0, stream>>>(h1, enc_W2, enc_b2, y2, 32);
        k_bn_relu<<<1, 256, 0, stream>>>(y2, enc_g2, enc_beta2, out + hoff[i], 128);
        // entropy of softmax(h, axis=1)
        k_row_entropy<<<4, 256, 0, stream>>>(out + hoff[i], acc + 1 + i);
        // projector: 128 -> 128 -> 64
        k_gemm_wmma<128, 128><<<1, 256, 0, stream>>>(out + hoff[i], prj_W1, prj_b1, y3, 32);
        k_bn_relu<<<1, 256, 0, stream>>>(y3, prj_g1, prj_beta1, h3, 128);
        k_gemm_wmma<64, 128><<<1, 256, 0, stream>>>(h3, prj_W2, prj_b2, y4, 32);
        k_bn_relu<<<1, 256, 0, stream>>>(y4, prj_g2, prj_beta2, out + zoff[i], 64);
    }

    k_finalize<<<1, 1, 0, stream>>>(acc, out + EL);
}